// FocalLoss_11003706212383
// MI455X (gfx1250) — compile-verified
//
#include <hip/hip_runtime.h>
#include <hip/hip_bf16.h>

#define BB 8
#define AA 120000
#define MM 64
#define CC 80

typedef __attribute__((ext_vector_type(4))) float f4;
typedef __attribute__((ext_vector_type(2))) float v2f;
typedef __attribute__((ext_vector_type(8))) float v8f;

// Exact f32 cross-wave (32 lane) sum using V_WMMA_F32_16X16X4_F32.
// A = ones(16x4)  (both A VGPRs = 1.0 in every lane -> all-ones matrix
// regardless of the (m,k)->(vgpr,lane) mapping).
// B = 4x16 with our 32 partials in VGPR0 and zeros in VGPR1: D[m,n] =
// sum_k B[k,n], so each 16-lane half of D's VGPR0 holds the 16 column
// sums; 4 xor-shuffles produce the total (= sum of all 64 B values =
// sum of the 32 partials) in every lane.
__device__ __forceinline__ float wave_sum_wmma(float x) {
  v2f a; a.x = 1.0f; a.y = 1.0f;
  v2f b; b.x = x;    b.y = 0.0f;
  v8f c = {};
  c = __builtin_amdgcn_wmma_f32_16x16x4_f32(
      /*neg_a=*/false, a, /*neg_b=*/false, b,
      /*c_mod=*/(short)0, c, /*reuse_a=*/false, /*reuse_b=*/false);
  float s = c[0];
  s += __shfl_xor(s, 1, 32);
  s += __shfl_xor(s, 2, 32);
  s += __shfl_xor(s, 4, 32);
  s += __shfl_xor(s, 8, 32);
  return s;
}

__global__ void fl_init(float* __restrict__ accf, int* __restrict__ acci) {
  int t = threadIdx.x;
  if (t < 16) accf[t] = 0.0f;   // [0..7] cls_sum, [8..15] reg_sum
  if (t < 16) acci[t] = 0;      // [0..7] n_sel,  [8..15] n_obj
}

__global__ __launch_bounds__(256)
void fl_main(const float* __restrict__ anchors,
             const float* __restrict__ regr,
             const float* __restrict__ cls,
             const float* __restrict__ ann,
             float* __restrict__ accf,
             int*   __restrict__ acci) {
  __shared__ f4 s_ann[MM * 2];     // [m*2]=box(x1,y1,x2,y2), [m*2+1]=(area,label,0,0)
  __shared__ float s_cls[8], s_reg[8];
  __shared__ int s_cnt[8];

  const int b   = (int)blockIdx.y;
  const int tid = (int)threadIdx.x;

  if (tid < MM) {
    const float* ap = ann + ((size_t)b * MM + tid) * 5;
    float x1 = ap[0], y1 = ap[1], x2 = ap[2], y2 = ap[3], lb = ap[4];
    f4 box; box.x = x1; box.y = y1; box.z = x2; box.w = y2;
    f4 mt;  mt.x = (x2 - x1) * (y2 - y1); mt.y = lb; mt.z = 0.0f; mt.w = 0.0f;
    s_ann[tid * 2]     = box;
    s_ann[tid * 2 + 1] = mt;
  }
  __syncthreads();

  const int a    = (int)blockIdx.x * 256 + tid;
  const bool live = (a < AA);
  const int ac   = live ? a : (AA - 1);          // clamp: keep EXEC uniform, no OOB
  const size_t base4 = ((size_t)b * AA + ac) * 4;

  f4 an = __builtin_nontemporal_load((const f4*)(anchors + base4));
  const float ax1 = an.x, ay1 = an.y, ax2 = an.z, ay2 = an.w;
  const float aArea = (ax2 - ax1) * (ay2 - ay1);

  // argmax IoU over annotations, tracked as rational (inter, union): no div.
  float bi = -1.0f, bu = 1.0f;   // encodes iou_max = -1 initially
  int bm = 0;
#pragma unroll 8
  for (int m = 0; m < MM; ++m) {
    f4 bx = s_ann[m * 2];
    f4 mt = s_ann[m * 2 + 1];
    float ix1 = fmaxf(ax1, bx.x);
    float iy1 = fmaxf(ay1, bx.y);
    float ix2 = fminf(ax2, bx.z);
    float iy2 = fminf(ay2, bx.w);
    float inter = fmaxf(ix2 - ix1, 0.0f) * fmaxf(iy2 - iy1, 0.0f);
    float uni   = fmaxf(aArea + mt.x - inter, 1e-8f);
    // valid && (inter/uni > bi/bu); strict '>' keeps first max (argmax semantics)
    bool upd = (mt.y != -1.0f) && (inter * bu > bi * uni);
    if (upd) { bi = inter; bu = uni; bm = m; }
  }

  bool obj = (bi > 0.5f * bu) && live;   // iou_max > IOU_OBJ
  bool bg  = (bi < 0.4f * bu);           // iou_max < IOU_BG
  bool sel = (obj || bg) && live;

  f4 gb = s_ann[bm * 2];
  f4 gm = s_ann[bm * 2 + 1];
  int lab = obj ? (int)gm.y : -1;

  // focal classification loss: sum_k alpha_t * p^2 * (-log p), p = t? c : 1-c
  float clsPart = 0.0f;
  if (sel) {
    const f4* cp = (const f4*)(cls + ((size_t)b * AA + ac) * CC);
#pragma unroll 5
    for (int j = 0; j < CC / 4; ++j) {
      f4 v = __builtin_nontemporal_load(cp + j);
      const int k = j * 4;
      {
        bool t = (k + 0 == lab);
        float p = t ? v.x : 1.0f - v.x;
        clsPart -= (t ? 0.25f : 0.75f) * p * p * __logf(p);
      }
      {
        bool t = (k + 1 == lab);
        float p = t ? v.y : 1.0f - v.y;
        clsPart -= (t ? 0.25f : 0.75f) * p * p * __logf(p);
      }
      {
        bool t = (k + 2 == lab);
        float p = t ? v.z : 1.0f - v.z;
        clsPart -= (t ? 0.25f : 0.75f) * p * p * __logf(p);
      }
      {
        bool t = (k + 3 == lab);
        float p = t ? v.w : 1.0f - v.w;
        clsPart -= (t ? 0.25f : 0.75f) * p * p * __logf(p);
      }
    }
  }

  // box regression loss for object anchors
  float regPart = 0.0f;
  if (obj) {
    float aw = ax2 - ax1, ah = ay2 - ay1;
    float acx = ax1 + 0.5f * aw, acy = ay1 + 0.5f * ah;
    float gw = fmaxf(gb.z - gb.x, 1.0f), gh = fmaxf(gb.w - gb.y, 1.0f);
    float gcx = gb.x + 0.5f * gw, gcy = gb.y + 0.5f * gh;
    f4 r = __builtin_nontemporal_load((const f4*)(regr + base4));
    float d0 = (gcx - acx) / aw - r.x;
    float d1 = (gcy - acy) / ah - r.y;
    float d2 = __logf(gw / aw) - r.z;
    float d3 = __logf(gh / ah) - r.w;
    regPart = d0 * d0 + d1 * d1 + d2 * d2 + d3 * d3;
  }

  int cnt = (sel ? 1 : 0) | (obj ? 0x10000 : 0);

  // wave-level exact f32 reduction via V_WMMA_F32_16X16X4_F32 (EXEC all-1s here)
  float wc = wave_sum_wmma(clsPart);
  float wr = wave_sum_wmma(regPart);
  cnt += __shfl_xor(cnt, 1, 32);
  cnt += __shfl_xor(cnt, 2, 32);
  cnt += __shfl_xor(cnt, 4, 32);
  cnt += __shfl_xor(cnt, 8, 32);
  cnt += __shfl_xor(cnt, 16, 32);

  const int lane = tid & 31, wave = tid >> 5;
  if (lane == 0) { s_cls[wave] = wc; s_reg[wave] = wr; s_cnt[wave] = cnt; }
  __syncthreads();
  if (tid == 0) {
    float tc = 0.0f, tr = 0.0f; int tn = 0;
#pragma unroll
    for (int w = 0; w < 8; ++w) { tc += s_cls[w]; tr += s_reg[w]; tn += s_cnt[w]; }
    atomicAdd(&accf[b],     tc);
    atomicAdd(&accf[8 + b], tr);
    atomicAdd(&acci[b],     tn & 0xFFFF);
    atomicAdd(&acci[8 + b], tn >> 16);
  }
}

__global__ void fl_final(const float* __restrict__ ann,
                         const float* __restrict__ accf,
                         const int*   __restrict__ acci,
                         float* __restrict__ out) {
  const int lane = (int)threadIdx.x;
  float clsL = 0.0f, regL = 0.0f;
  if (lane < BB) {
    bool has = false;
    for (int m = 0; m < MM; ++m)
      has = has || (ann[((size_t)lane * MM + m) * 5 + 4] != -1.0f);
    int ns = acci[lane];      if (ns < 1) ns = 1;
    int no = acci[8 + lane];
    if (has) clsL = accf[lane] / (float)ns;
    int dn = no * 4;          if (dn < 1) dn = 1;
    if (has && no > 0) regL = accf[8 + lane] / (float)dn;
  }
  for (int off = 16; off >= 1; off >>= 1) {
    clsL += __shfl_xor(clsL, off, 32);
    regL += __shfl_xor(regL, off, 32);
  }
  if (lane == 0) {
    out[0] = clsL / (float)BB;
    out[1] = regL / (float)BB;
  }
}

extern "C" void kernel_launch(void* const* d_in, const int* in_sizes, int n_in,
                              void* d_out, int out_size, void* d_ws, size_t ws_size,
                              hipStream_t stream) {
  (void)in_sizes; (void)n_in; (void)out_size; (void)ws_size;
  const float* anchors = (const float*)d_in[0];  // (B,A,4)
  const float* regr    = (const float*)d_in[1];  // (B,A,4)
  const float* cls     = (const float*)d_in[2];  // (B,A,C)
  const float* ann     = (const float*)d_in[3];  // (B,M,5)
  float* accf = (float*)d_ws;                    // 16 floats
  int*   acci = (int*)((char*)d_ws + 64);        // 16 ints
  float* out  = (float*)d_out;

  fl_init<<<1, 64, 0, stream>>>(accf, acci);
  dim3 grid((AA + 255) / 256, BB);
  fl_main<<<grid, 256, 0, stream>>>(anchors, regr, cls, ann, accf, acci);
  fl_final<<<1, 32, 0, stream>>>(ann, accf, acci, out);
}